// NegativeBinomialRegressionModel_26190710571426
// MI455X (gfx1250) — compile-verified
//
#include <hip/hip_runtime.h>
#include <hip/hip_bf16.h>
#include <math.h>

#define N_SAMP 1024
#define P_DIM  4
#define G_GENES 20000

typedef float v2f __attribute__((ext_vector_type(2)));
typedef float v8f __attribute__((ext_vector_type(8)));

__device__ __forceinline__ float softplusf(float x) {
    return fmaxf(x, 0.0f) + log1pf(expf(-fabsf(x)));
}

// Branch-free lgamma for x > 0: shift x into [8, inf) via the recurrence
// lgamma(x) = lgamma(x+n) - log(prod_{i<n}(x+i)), then Stirling.
// Uses hardware v_log_f32 (2 calls total).
__device__ __forceinline__ float lgamma_pos(float x) {
    float xx = x;
    float p = 1.0f;
    #pragma unroll
    for (int i = 0; i < 8; ++i) {          // from any x>0 this reaches xx>=8
        bool sh = xx < 8.0f;
        p  = sh ? p * xx : p;
        xx = sh ? xx + 1.0f : xx;
    }
    const float lx   = __logf(xx);
    const float inv  = 1.0f / xx;
    const float inv2 = inv * inv;
    // 1/(12x) - 1/(360x^3) + 1/(1260x^5)
    const float series = ((7.93650794e-4f * inv2 - 2.77777778e-3f) * inv2
                          + 8.33333333e-2f) * inv;
    return (xx - 0.5f) * lx - xx + 0.918938533204672f + series - __logf(p);
}

// ---------------- K0: zero the scalar output ----------------
__global__ void k0_zero(float* out) {
    if (threadIdx.x == 0) out[0] = 0.0f;
}

// ---------------- K1: per-gene precompute (packed) + priors ----------------
// pg4[g] = (r, log r, lgamma r, mu) ; bt4[g] = beta[0..3][g]
__global__ __launch_bounds__(256) void k1_pergene(
    const float* __restrict__ mu, const float* __restrict__ beta,
    const float* __restrict__ phi,
    float4* __restrict__ pg4, float4* __restrict__ bt4,
    float* __restrict__ out)
{
    const float M0c = 0.0f, S0c = 2.0f, SHAPEc = 2.0f, SCALEc = 1.0f;
    const int g = blockIdx.x * blockDim.x + threadIdx.x;
    float acc = 0.0f;
    if (g < G_GENES) {
        const float sp = softplusf(phi[g]);
        const float r  = 1.0f / sp;
        const float muv = mu[g];
        pg4[g] = make_float4(r, __logf(r), lgamma_pos(r), muv);
        const float b0 = beta[0 * G_GENES + g], b1 = beta[1 * G_GENES + g];
        const float b2 = beta[2 * G_GENES + g], b3 = beta[3 * G_GENES + g];
        bt4[g] = make_float4(b0, b1, b2, b3);
        // normal priors on mu and beta
        const float c0 = -0.5f * logf(2.0f * (float)M_PI * S0c * S0c);
        const float inv2s2 = 1.0f / (2.0f * S0c * S0c);
        float dmu = muv - M0c;
        acc += c0 - dmu * dmu * inv2s2;
        acc += c0 - (b0 - M0c) * (b0 - M0c) * inv2s2;
        acc += c0 - (b1 - M0c) * (b1 - M0c) * inv2s2;
        acc += c0 - (b2 - M0c) * (b2 - M0c) * inv2s2;
        acc += c0 - (b3 - M0c) * (b3 - M0c) * inv2s2;
        // gamma prior on softplus(phi)
        acc += (SHAPEc - 1.0f) * logf(sp) - sp / SCALEc
             - lgammaf(SHAPEc) - SHAPEc * logf(SCALEc);
    }
    __shared__ float red[256];
    red[threadIdx.x] = acc;
    __syncthreads();
    for (int s = 128; s > 0; s >>= 1) {
        if (threadIdx.x < s) red[threadIdx.x] += red[threadIdx.x + s];
        __syncthreads();
    }
    if (threadIdx.x == 0) atomicAdd(out, red[0]);
}

// ---------------- K2: WMMA 16x16x4 f32, two-pass logsumexp ----------------
// One block per 16-row stripe; 8 waves; pass 1 = max only, pass 2 = sum-exp.
__global__ __launch_bounds__(256) void k2_logsumexp(
    const float* __restrict__ X, const float* __restrict__ beta,
    const float* __restrict__ mu, float* __restrict__ norm_ws)
{
    const int tid  = threadIdx.x;
    const int wave = tid >> 5;
    const int lane = tid & 31;
    const int half = lane >> 4;    // 0: K=0..1, 1: K=2..3
    const int l16  = lane & 15;
    const int rowBase = blockIdx.x * 16;

    // A tile (16x4 f32)
    v2f a;
    {
        const float* xp = X + (rowBase + l16) * P_DIM + half * 2;
        a[0] = xp[0];
        a[1] = xp[1];
    }

    __shared__ float buf[8 * 256];
    __shared__ float rowMaxSh[16];
    const int nTiles = G_GENES / 16;   // 1250 exact

    // ---- pass 1: row maxima (no exp) ----
    float runM[8];
    #pragma unroll
    for (int v = 0; v < 8; ++v) runM[v] = -INFINITY;

    for (int t = wave; t < nTiles; t += 8) {   // wave-uniform -> EXEC all-1s
        const int gl = t * 16 + l16;
        v2f b;
        b[0] = beta[(half * 2 + 0) * G_GENES + gl];
        b[1] = beta[(half * 2 + 1) * G_GENES + gl];
        const float muv = mu[gl];
        v8f c;
        #pragma unroll
        for (int v = 0; v < 8; ++v) c[v] = muv;
        v8f d = __builtin_amdgcn_wmma_f32_16x16x4_f32(
            false, a, false, b, (short)0, c, false, false);
        #pragma unroll
        for (int v = 0; v < 8; ++v) runM[v] = fmaxf(runM[v], d[v]);
    }
    #pragma unroll
    for (int v = 0; v < 8; ++v) buf[wave * 256 + v * 32 + lane] = runM[v];
    __syncthreads();
    if (tid < 16) {
        const int v       = (tid < 8) ? tid : tid - 8;
        const int laneOff = (tid < 8) ? 0 : 16;
        float M = -INFINITY;
        for (int w = 0; w < 8; ++w)
            for (int l = 0; l < 16; ++l)
                M = fmaxf(M, buf[w * 256 + v * 32 + laneOff + l]);
        rowMaxSh[tid] = M;
    }
    __syncthreads();

    // ---- pass 2: sum of exp(z - rowmax) ----
    float rmax[8], runS[8];
    #pragma unroll
    for (int v = 0; v < 8; ++v) {
        rmax[v] = rowMaxSh[half ? v + 8 : v];
        runS[v] = 0.0f;
    }
    for (int t = wave; t < nTiles; t += 8) {
        const int gl = t * 16 + l16;
        v2f b;
        b[0] = beta[(half * 2 + 0) * G_GENES + gl];
        b[1] = beta[(half * 2 + 1) * G_GENES + gl];
        const float muv = mu[gl];
        v8f c;
        #pragma unroll
        for (int v = 0; v < 8; ++v) c[v] = muv;
        v8f d = __builtin_amdgcn_wmma_f32_16x16x4_f32(
            false, a, false, b, (short)0, c, false, false);
        #pragma unroll
        for (int v = 0; v < 8; ++v) runS[v] += __expf(d[v] - rmax[v]);
    }
    #pragma unroll
    for (int v = 0; v < 8; ++v) buf[wave * 256 + v * 32 + lane] = runS[v];
    __syncthreads();
    if (tid < 16) {
        const int v       = (tid < 8) ? tid : tid - 8;
        const int laneOff = (tid < 8) ? 0 : 16;
        float S = 0.0f;
        for (int w = 0; w < 8; ++w)
            for (int l = 0; l < 16; ++l)
                S += buf[w * 256 + v * 32 + laneOff + l];
        norm_ws[rowBase + tid] = rowMaxSh[tid] + __logf(S);
    }
}

// ---------------- K3: NB log-likelihood, one block per row ----------------
__global__ __launch_bounds__(256) void k3_loglik(
    const float* __restrict__ X, const float* __restrict__ Y,
    const float4* __restrict__ pg4, const float4* __restrict__ bt4,
    const float* __restrict__ norm_ws, float* __restrict__ out)
{
    const int row = blockIdx.x;
    const int tid = threadIdx.x;
    __shared__ float red[256];
    __shared__ float lgY[512];   // lgamma(y+1), y integer in [0,500)
    __shared__ float sh_s;

    for (int k = tid; k < 512; k += 256) lgY[k] = lgammaf((float)k + 1.0f);

    // library size s[row]
    const float* Yr = Y + (size_t)row * G_GENES;
    const float4* Y4 = (const float4*)Yr;
    float ps = 0.0f;
    for (int i = tid; i < G_GENES / 4; i += 256) {
        float4 y4 = Y4[i];
        ps += y4.x + y4.y + y4.z + y4.w;
    }
    red[tid] = ps;
    __syncthreads();
    for (int s = 128; s > 0; s >>= 1) {
        if (tid < s) red[tid] += red[tid + s];
        __syncthreads();
    }
    if (tid == 0) sh_s = red[0];
    __syncthreads();

    const float logs = __logf(sh_s);
    const float nrm  = norm_ws[row];
    const float x0 = X[row * 4 + 0], x1 = X[row * 4 + 1];
    const float x2 = X[row * 4 + 2], x3 = X[row * 4 + 3];

    float acc = 0.0f;
    for (int g = tid; g < G_GENES; g += 256) {
        const float  y  = Yr[g];
        const float4 pg = pg4[g];   // r, log r, lgamma r, mu
        const float4 bt = bt4[g];   // beta columns
        float z = pg.w + x0 * bt.x + x1 * bt.y + x2 * bt.z + x3 * bt.w;
        const float lm   = logs + (z - nrm);   // log(mean)
        const float mean = __expf(lm);
        const float lrm  = __logf(pg.x + mean);
        int yi = (int)y; if (yi > 511) yi = 511;
        acc += lgamma_pos(y + pg.x) - pg.z - lgY[yi]
             + pg.x * (pg.y - lrm) + y * (lm - lrm);
    }

    __syncthreads();
    red[tid] = acc;
    __syncthreads();
    for (int s = 128; s > 0; s >>= 1) {
        if (tid < s) red[tid] += red[tid + s];
        __syncthreads();
    }
    if (tid == 0) atomicAdd(out, red[0]);
}

extern "C" void kernel_launch(void* const* d_in, const int* in_sizes, int n_in,
                              void* d_out, int out_size, void* d_ws, size_t ws_size,
                              hipStream_t stream)
{
    const float* X    = (const float*)d_in[0];
    const float* Y    = (const float*)d_in[1];
    const float* mu   = (const float*)d_in[2];
    const float* beta = (const float*)d_in[3];
    const float* phi  = (const float*)d_in[4];
    float* out = (float*)d_out;

    float*  ws      = (float*)d_ws;
    float4* pg4     = (float4*)ws;                   // G float4
    float4* bt4     = (float4*)(ws + 4 * G_GENES);   // G float4
    float*  norm_ws = ws + 8 * G_GENES;              // N floats (~645 KB total)

    k0_zero<<<1, 32, 0, stream>>>(out);
    k1_pergene<<<(G_GENES + 255) / 256, 256, 0, stream>>>(
        mu, beta, phi, pg4, bt4, out);
    k2_logsumexp<<<N_SAMP / 16, 256, 0, stream>>>(X, beta, mu, norm_ws);
    k3_loglik<<<N_SAMP, 256, 0, stream>>>(X, Y, pg4, bt4, norm_ws, out);
}